// DKVMN_515396075525
// MI455X (gfx1250) — compile-verified
//
#include <hip/hip_runtime.h>
#include <hip/hip_bf16.h>

typedef float v2f __attribute__((ext_vector_type(2)));
typedef float v8f __attribute__((ext_vector_type(8)));

#define WMMA_F32(a, b, c) \
    __builtin_amdgcn_wmma_f32_16x16x4_f32(false, (a), false, (b), (short)0, (c), false, false)

constexpr int B_ = 32;
constexpr int S_ = 2048;
constexpr int D_ = 64;
constexpr int M_ = 32;

// ---------------------------------------------------------------------------
// Phase 1: fused gather + w/e/a.  One wave per 16-row tile (rows = b*S+s).
//   w  = softmax(k @ Mk^T)           [rows,32]
//   e  = sigmoid(v @ e_W + e_b)      [rows,64]
//   a  = tanh   (v @ a_W + a_b)      [rows,64]
// f32 WMMA 16x16x4, K=64 -> 16 chained WMMAs per 16x16 tile.
// ---------------------------------------------------------------------------
__global__ __launch_bounds__(256) void wea_kernel(
    const int* __restrict__ skill, const int* __restrict__ response,
    const float* __restrict__ k_emb, const float* __restrict__ v_emb,
    const float* __restrict__ Mk,
    const float* __restrict__ eW, const float* __restrict__ eb,
    const float* __restrict__ aW, const float* __restrict__ ab,
    float* __restrict__ w_ws, float* __restrict__ e_ws, float* __restrict__ a_ws,
    int num_skill)
{
    __shared__ float lgt[8][16][33];    // per-wave softmax scratch (padded)

    const int tid  = threadIdx.x;
    const int lane = tid & 31;
    const int wave = tid >> 5;
    const int tile = blockIdx.x * 8 + wave;      // 4096 tiles total
    const int rowBase = tile * 16;

    const int r = lane & 15;                     // tile row handled by this lane
    const int h = lane >> 4;                     // K-half select (WMMA f32 A layout)
    const int row = rowBase + r;

    const int sk = skill[row];
    const int xc = sk + num_skill * response[row];
    const float* kRow = k_emb + (long)sk * 64;
    const float* vRow = v_emb + (long)xc * 64;

    // ---- w logits: (16x64) @ B[64x32], B[kr][n] = Mk[n][kr]
    v8f c0 = {}, c1 = {};
#pragma unroll
    for (int kk = 0; kk < 16; ++kk) {
        const int bk = kk * 4 + 2 * h;
        v2f av; av.x = kRow[bk]; av.y = kRow[bk + 1];
        v2f b0; b0.x = Mk[r * 64 + bk];        b0.y = Mk[r * 64 + bk + 1];
        v2f b1; b1.x = Mk[(r + 16) * 64 + bk]; b1.y = Mk[(r + 16) * 64 + bk + 1];
        c0 = WMMA_F32(av, b0, c0);
        c1 = WMMA_F32(av, b1, c1);
    }

    // stash logits in per-wave LDS region, then row-wise softmax by 16 lanes
#pragma unroll
    for (int j = 0; j < 8; ++j) {
        const int rr = j + 8 * h;
        lgt[wave][rr][r]      = c0[j];
        lgt[wave][rr][r + 16] = c1[j];
    }
    __syncthreads();

    if (lane < 16) {
        float mx = -1e30f;
#pragma unroll
        for (int m = 0; m < 32; ++m) mx = fmaxf(mx, lgt[wave][lane][m]);
        float sum = 0.f;
#pragma unroll
        for (int m = 0; m < 32; ++m) sum += __expf(lgt[wave][lane][m] - mx);
        const float inv = 1.0f / sum;
        float* wout = w_ws + (long)(rowBase + lane) * 32;
#pragma unroll
        for (int m = 0; m < 32; ++m) wout[m] = __expf(lgt[wave][lane][m] - mx) * inv;
    }

    // ---- e and a: (16x64) @ (64x64), four 16-col tiles each
    v8f ce[4] = {{}, {}, {}, {}};
    v8f ca[4] = {{}, {}, {}, {}};
#pragma unroll
    for (int kk = 0; kk < 16; ++kk) {
        const int bk = kk * 4 + 2 * h;
        v2f av; av.x = vRow[bk]; av.y = vRow[bk + 1];
#pragma unroll
        for (int nt = 0; nt < 4; ++nt) {
            const int n = r + nt * 16;
            v2f be; be.x = eW[bk * 64 + n]; be.y = eW[(bk + 1) * 64 + n];
            ce[nt] = WMMA_F32(av, be, ce[nt]);
            v2f ba; ba.x = aW[bk * 64 + n]; ba.y = aW[(bk + 1) * 64 + n];
            ca[nt] = WMMA_F32(av, ba, ca[nt]);
        }
    }
#pragma unroll
    for (int nt = 0; nt < 4; ++nt) {
        const int n = r + nt * 16;
        const float ebn = eb[n];
        const float abn = ab[n];
#pragma unroll
        for (int j = 0; j < 8; ++j) {
            const int rr = j + 8 * h;
            const long o = (long)(rowBase + rr) * 64 + n;
            const float ve = ce[nt][j] + ebn;
            e_ws[o] = 1.0f / (1.0f + __expf(-ve));
            a_ws[o] = tanhf(ca[nt][j] + abn);
        }
    }
}

// ---------------------------------------------------------------------------
// Phase 2: sequential memory-state scan, one block per batch, barrier-free.
// tid = d*4 + mg; thread owns Mv[mg*8 .. mg*8+7][d] in registers.
//   Mv <- Mv*(1 - w*e) + w*a   (mask folded into w for the update)
//   read[t] = sum_m w[t+1][m] * Mv[m][:]   (shfl_xor reduce over 4 lanes)
// Loads are software-pipelined one step ahead (w two steps ahead).
// ---------------------------------------------------------------------------
__global__ __launch_bounds__(256) void scan_kernel(
    const float* __restrict__ w_ws, const float* __restrict__ e_ws,
    const float* __restrict__ a_ws, const float* __restrict__ mask,
    const float* __restrict__ Mv0, float* __restrict__ read_ws)
{
    const int b   = blockIdx.x;
    const int tid = threadIdx.x;
    const int d   = tid >> 2;
    const int mg  = tid & 3;

    const float* wp = w_ws + (long)b * S_ * 32 + mg * 8;
    const float* ep = e_ws + (long)b * S_ * 64 + d;
    const float* ap = a_ws + (long)b * S_ * 64 + d;
    const float* mp = mask + (long)b * S_;
    float* rp = read_ws + (long)b * (S_ - 1) * 64 + d;

    float mv[8];
#pragma unroll
    for (int i = 0; i < 8; ++i) mv[i] = Mv0[(mg * 8 + i) * 64 + d];

    float wcur[8], wnxt[8];
#pragma unroll
    for (int i = 0; i < 8; ++i) wcur[i] = wp[i];
#pragma unroll
    for (int i = 0; i < 8; ++i) wnxt[i] = wp[32 + i];
    float ed = ep[0], ad = ap[0], mk = mp[0];

#pragma unroll 2
    for (int t = 0; t < S_; ++t) {
        const int t1 = (t + 1 < S_) ? t + 1 : S_ - 1;
        const int t2 = (t + 2 < S_) ? t + 2 : S_ - 1;

        // prefetch next step's operands
        float wfut[8];
#pragma unroll
        for (int i = 0; i < 8; ++i) wfut[i] = wp[(long)t2 * 32 + i];
        const float ed2 = ep[(long)t1 * 64];
        const float ad2 = ap[(long)t1 * 64];
        const float mk2 = mp[t1];

        // update + read accumulation
        float racc = 0.f;
#pragma unroll
        for (int i = 0; i < 8; ++i) {
            const float wm = wcur[i] * mk;
            const float t0 = wm * ed;
            mv[i] = fmaf(-t0, mv[i], mv[i]);
            mv[i] = fmaf(wm, ad, mv[i]);
            racc  = fmaf(wnxt[i], mv[i], racc);   // w[t+1] . Mv (post-update)
        }
        racc += __shfl_xor(racc, 1, 32);
        racc += __shfl_xor(racc, 2, 32);
        if (mg == 0 && t < S_ - 1) rp[(long)t * 64] = racc;

#pragma unroll
        for (int i = 0; i < 8; ++i) { wcur[i] = wnxt[i]; wnxt[i] = wfut[i]; }
        ed = ed2; ad = ad2; mk = mk2;
    }
}

// ---------------------------------------------------------------------------
// Phase 3: f = tanh([read, k(:,1:)] @ f_W + f_b);  p = f @ p_W + p_b
// One wave per 16-row tile (rows = b*(S-1)+s), K=128 -> 32 chained WMMAs
// per 16x16 tile, 4 N-tiles.  k is re-gathered from the embedding table.
// ---------------------------------------------------------------------------
__global__ __launch_bounds__(256) void fp_kernel(
    const int* __restrict__ skill, const float* __restrict__ k_emb,
    const float* __restrict__ read_ws,
    const float* __restrict__ fW, const float* __restrict__ fb,
    const float* __restrict__ pW, const float* __restrict__ pb,
    float* __restrict__ out)
{
    __shared__ float s_pw[64];
    __shared__ float ftile[8][16][65];

    const int tid = threadIdx.x;
    if (tid < 64) s_pw[tid] = pW[tid];
    __syncthreads();

    const int lane = tid & 31;
    const int wave = tid >> 5;
    const int totalRows = B_ * (S_ - 1);           // 65504 = 16*4094
    int tile = blockIdx.x * 8 + wave;
    bool valid = (tile * 16 < totalRows);
    if (!valid) tile = 0;                          // clamp; no early return (barriers)
    const int rowBase = tile * 16;

    const int r = lane & 15;
    const int h = lane >> 4;
    const int row = rowBase + r;
    const int b = row / (S_ - 1);
    const int s = row - b * (S_ - 1);

    const float* rdRow = read_ws + (long)row * 64;
    const int sk = skill[b * S_ + s + 1];
    const float* kRow = k_emb + (long)sk * 64;

    v8f cf[4] = {{}, {}, {}, {}};
#pragma unroll
    for (int kk = 0; kk < 16; ++kk) {              // K = 0..63 : read part
        const int bk = kk * 4 + 2 * h;
        v2f av; av.x = rdRow[bk]; av.y = rdRow[bk + 1];
#pragma unroll
        for (int nt = 0; nt < 4; ++nt) {
            const int n = r + nt * 16;
            v2f bv; bv.x = fW[bk * 64 + n]; bv.y = fW[(bk + 1) * 64 + n];
            cf[nt] = WMMA_F32(av, bv, cf[nt]);
        }
    }
#pragma unroll
    for (int kk = 0; kk < 16; ++kk) {              // K = 64..127 : k part
        const int bk = kk * 4 + 2 * h;
        const int bk2 = bk + 64;
        v2f av; av.x = kRow[bk]; av.y = kRow[bk + 1];
#pragma unroll
        for (int nt = 0; nt < 4; ++nt) {
            const int n = r + nt * 16;
            v2f bv; bv.x = fW[bk2 * 64 + n]; bv.y = fW[(bk2 + 1) * 64 + n];
            cf[nt] = WMMA_F32(av, bv, cf[nt]);
        }
    }

    // bias + tanh -> per-wave LDS tile
#pragma unroll
    for (int nt = 0; nt < 4; ++nt) {
        const int n = r + nt * 16;
        const float fbn = fb[n];
#pragma unroll
        for (int j = 0; j < 8; ++j) {
            const int rr = j + 8 * h;
            ftile[wave][rr][n] = tanhf(cf[nt][j] + fbn);
        }
    }
    __syncthreads();

    if (lane < 16) {
        const int orow = rowBase + lane;
        if (valid && orow < totalRows) {
            float acc = pb[0];
#pragma unroll
            for (int n = 0; n < 64; ++n) acc = fmaf(ftile[wave][lane][n], s_pw[n], acc);
            out[orow] = acc;
        }
    }
}

// ---------------------------------------------------------------------------
extern "C" void kernel_launch(void* const* d_in, const int* in_sizes, int n_in,
                              void* d_out, int out_size, void* d_ws, size_t ws_size,
                              hipStream_t stream)
{
    (void)n_in; (void)out_size; (void)ws_size;
    const int*   response = (const int*)d_in[1];
    const int*   skill    = (const int*)d_in[2];
    const float* mask     = (const float*)d_in[3];
    const float* k_emb    = (const float*)d_in[4];
    const float* v_emb    = (const float*)d_in[5];
    const float* Mk       = (const float*)d_in[6];
    const float* Mv0      = (const float*)d_in[7];
    const float* eW       = (const float*)d_in[8];
    const float* eb       = (const float*)d_in[9];
    const float* aW       = (const float*)d_in[10];
    const float* ab       = (const float*)d_in[11];
    const float* fW       = (const float*)d_in[12];
    const float* fb       = (const float*)d_in[13];
    const float* pW       = (const float*)d_in[14];
    const float* pb       = (const float*)d_in[15];
    const int num_skill   = in_sizes[4] / 64;

    float* w_ws    = (float*)d_ws;                         // B*S*32
    float* e_ws    = w_ws + (size_t)B_ * S_ * 32;          // B*S*64
    float* a_ws    = e_ws + (size_t)B_ * S_ * 64;          // B*S*64
    float* read_ws = a_ws + (size_t)B_ * S_ * 64;          // B*(S-1)*64

    const int tiles1 = (B_ * S_) / 16;                     // 4096
    wea_kernel<<<tiles1 / 8, 256, 0, stream>>>(
        skill, response, k_emb, v_emb, Mk, eW, eb, aW, ab,
        w_ws, e_ws, a_ws, num_skill);

    scan_kernel<<<B_, 256, 0, stream>>>(w_ws, e_ws, a_ws, mask, Mv0, read_ws);

    const int tiles3 = (B_ * (S_ - 1) + 15) / 16;          // 4094
    fp_kernel<<<(tiles3 + 7) / 8, 256, 0, stream>>>(
        skill, k_emb, read_ws, fW, fb, pW, pb, (float*)d_out);
}